// QuantumLayer_81458349736031
// MI455X (gfx1250) — compile-verified
//
#include <hip/hip_runtime.h>

typedef float v2f __attribute__((ext_vector_type(2)));
typedef float v8f __attribute__((ext_vector_type(8)));

#define WMMA4(A, B, C) \
  __builtin_amdgcn_wmma_f32_16x16x4_f32(false, (A), false, (B), (short)0, (C), false, false)

static __device__ __forceinline__ v2f mk2(float a, float b) {
  v2f r; r[0] = a; r[1] = b; return r;
}

// ---------------------------------------------------------------------------
// Pre-kernel: build the 4x2 complex 16x16 gate matrices with the CNOT-chain
// permutations folded in, stored directly in WMMA A-operand layout.
//   ws layout (floats): [layer][gate(0=row/q0-3,1=col/q4-7)][part(re,im)][lane(32)][8]
// A-layout for f32 16x16x4 slice j: element (m,k), k=4j+p+2s -> lane m+16s, dword j*2+p.
// ---------------------------------------------------------------------------
__global__ void __launch_bounds__(256)
qprep_gates_81458349(const float* __restrict__ w, float* __restrict__ gws) {
  const int t  = threadIdx.x;          // 256 threads: one (row, col) element each
  const int r  = t >> 4;
  const int cc = t & 15;
  // CNOT chain inside a nibble = prefix-xor permutation of rows
  const int sr   = (r ^ (r >> 1) ^ (r >> 2) ^ (r >> 3)) & 15;
  const int lane = sr + 16 * ((cc & 3) >> 1);
  const int dw   = (cc >> 2) * 2 + (cc & 1);
  for (int l = 0; l < 4; ++l) {
    for (int g = 0; g < 2; ++g) {
      float re = 1.f, im = 0.f;
      #pragma unroll
      for (int qq = 0; qq < 4; ++qq) {
        const int q = g * 4 + qq;                // qubit index
        const float th = w[(l * 8 + q) * 2 + 0];
        const float tz = w[(l * 8 + q) * 2 + 1];
        const float c  = cosf(0.5f * th), s = sinf(0.5f * th);
        const float pc = cosf(0.5f * tz), ps = sinf(0.5f * tz);
        // qubit q maps to bit (3-qq) of the nibble index (qubit 0 = MSB)
        const int bi = (r  >> (3 - qq)) & 1;
        const int bj = (cc >> (3 - qq)) & 1;
        const float ry  = bi ? (bj ? c : s) : (bj ? -s : c);   // RY[bi][bj]
        const float phi = bi ? ps : -ps;                        // e^{-+ i tz/2}
        const float gr = ry * pc, gi = ry * phi;
        const float nre = re * gr - im * gi;
        const float nim = re * gi + im * gr;
        re = nre; im = nim;
      }
      const int base = ((l * 2 + g) * 2) * 256;
      gws[base +       lane * 8 + dw] = re;
      gws[base + 256 + lane * 8 + dw] = im;
    }
  }
}

// ---------------------------------------------------------------------------
// Main kernel: one wave per batch element. State = 16x16 complex matrix
// (rows = qubits 0-3 nibble, cols = qubits 4-7 nibble).
// Per layer: Tr/Ti = U_row' x S  (state as B operand), LDS transpose trip,
//            O = U_col' x T^T    (stored transposed), CNOT(3,4) fix-up folded
//            into the next read's addressing (c -> c^15 on odd rows).
// ---------------------------------------------------------------------------
__global__ void __launch_bounds__(256)
qsim_wmma_81458349(const float* __restrict__ x, const float* __restrict__ gates,
                   float* __restrict__ out, int B) {
  __shared__ float lds[8 * 512];                 // 2KB per wave: [0..255]=re, [256..511]=im
  const int lane = threadIdx.x & 31;
  const int wid  = threadIdx.x >> 5;
  const int b    = blockIdx.x * 8 + wid;
  if (b >= B) return;                            // wave-uniform exit: EXEC all-1 inside
  float* L = &lds[wid * 512];
  const int n = lane & 15;                       // B/C/D column index
  const int s = lane >> 4;                       // lane half

  // ---- encoding: product state amplitudes -------------------------------
  // |arg| <= pi/2, so the hardware v_sin/v_cos path is accurate enough vs a
  // float32 reference; precise libm trig here costs more VALU than the WMMAs.
  float cs[8], sn[8];
  #pragma unroll
  for (int q = 0; q < 8; ++q) {
    const float a = 0.5f * x[(size_t)b * 8 + q];
    cs[q] = __cosf(a); sn[q] = __sinf(a);
  }
  float aR[16];
  #pragma unroll
  for (int i = 0; i < 16; ++i) {
    float v = ((i >> 3) & 1) ? sn[0] : cs[0];
    v *= ((i >> 2) & 1) ? sn[1] : cs[1];
    v *= ((i >> 1) & 1) ? sn[2] : cs[2];
    v *= ((i     ) & 1) ? sn[3] : cs[3];
    aR[i] = v;
  }
  const float aCl = (((n >> 3) & 1) ? sn[4] : cs[4]) *
                    (((n >> 2) & 1) ? sn[5] : cs[5]) *
                    (((n >> 1) & 1) ? sn[6] : cs[6]) *
                    (((n     ) & 1) ? sn[7] : cs[7]);

  // Initial state directly in B-operand layout (real; imag = 0)
  v2f Br[4], Bi[4];
  #pragma unroll
  for (int j = 0; j < 4; ++j) {
    Br[j] = mk2((s ? aR[4 * j + 2] : aR[4 * j + 0]) * aCl,
                (s ? aR[4 * j + 3] : aR[4 * j + 1]) * aCl);
    Bi[j] = mk2(0.f, 0.f);
  }

  const float4* g4 = (const float4*)gates;
  // LDS bases (floats): column-major image  addr = col*16 + row
  const int wb   = n * 16 + 8 * s;               // C/D write base
  const int rb0  = (2 * s) * 16 + n;             // read base, p=0 (both trips)
  const int rb1p = (2 * s + 1) * 16 + n;         // trip-1 read, p=1 (plain)
  const int rb1f = (2 * s + 1) * 16 + (15 - n);  // trip-2 read, p=1 (c^15 fix-up)

  v8f Or = {}, Oi = {};
  #pragma unroll
  for (int l = 0; l < 4; ++l) {
    // gate A operands (hit L1: same 16KB for every wave)
    v2f ARr[4], ARi[4], ACr[4], ACi[4];
    {
      const int f0 = ((l * 2 + 0) * 2 + 0) * 64 + lane * 2;
      const int f1 = ((l * 2 + 0) * 2 + 1) * 64 + lane * 2;
      const int f2 = ((l * 2 + 1) * 2 + 0) * 64 + lane * 2;
      const int f3 = ((l * 2 + 1) * 2 + 1) * 64 + lane * 2;
      float4 a, c2;
      a = g4[f0]; c2 = g4[f0 + 1];
      ARr[0] = mk2(a.x, a.y); ARr[1] = mk2(a.z, a.w); ARr[2] = mk2(c2.x, c2.y); ARr[3] = mk2(c2.z, c2.w);
      a = g4[f1]; c2 = g4[f1 + 1];
      ARi[0] = mk2(a.x, a.y); ARi[1] = mk2(a.z, a.w); ARi[2] = mk2(c2.x, c2.y); ARi[3] = mk2(c2.z, c2.w);
      a = g4[f2]; c2 = g4[f2 + 1];
      ACr[0] = mk2(a.x, a.y); ACr[1] = mk2(a.z, a.w); ACr[2] = mk2(c2.x, c2.y); ACr[3] = mk2(c2.z, c2.w);
      a = g4[f3]; c2 = g4[f3 + 1];
      ACi[0] = mk2(a.x, a.y); ACi[1] = mk2(a.z, a.w); ACi[2] = mk2(c2.x, c2.y); ACi[3] = mk2(c2.z, c2.w);
    }

    // ---- row-gate multiply: T = U_row' x S ------------------------------
    v8f Tr, Ti;
    if (l == 0) {                                // encoded state is real
      v8f m1 = {}, m4 = {};
      #pragma unroll
      for (int j = 0; j < 4; ++j) {
        m1 = WMMA4(ARr[j], Br[j], m1);
        m4 = WMMA4(ARi[j], Br[j], m4);
      }
      Tr = m1; Ti = m4;
    } else {
      // read W in B layout (contract rows) with CNOT(3,4) fix-up on odd rows
      #pragma unroll
      for (int j = 0; j < 4; ++j) {
        Br[j] = mk2(L[rb0 + 64 * j],       L[rb1f + 64 * j]);
        Bi[j] = mk2(L[256 + rb0 + 64 * j], L[256 + rb1f + 64 * j]);
      }
      v8f m1 = {}, m2 = {}, m3 = {}, m4 = {};
      #pragma unroll
      for (int j = 0; j < 4; ++j) {
        m1 = WMMA4(ARr[j], Br[j], m1);
        m2 = WMMA4(ARi[j], Bi[j], m2);
        m3 = WMMA4(ARr[j], Bi[j], m3);
        m4 = WMMA4(ARi[j], Br[j], m4);
      }
      Tr = m1 - m2; Ti = m3 + m4;
    }

    // ---- LDS transpose trip: write T (C/D column-major), read T^T as B --
    #pragma unroll
    for (int v = 0; v < 8; ++v) {
      L[wb + v]       = Tr[v];
      L[256 + wb + v] = Ti[v];
    }
    #pragma unroll
    for (int j = 0; j < 4; ++j) {
      Br[j] = mk2(L[rb0 + 64 * j],       L[rb1p + 64 * j]);
      Bi[j] = mk2(L[256 + rb0 + 64 * j], L[256 + rb1p + 64 * j]);
    }

    // ---- col-gate multiply: O^T = U_col' x T^T --------------------------
    {
      v8f m1 = {}, m2 = {}, m3 = {}, m4 = {};
      #pragma unroll
      for (int j = 0; j < 4; ++j) {
        m1 = WMMA4(ACr[j], Br[j], m1);
        m2 = WMMA4(ACi[j], Bi[j], m2);
        m3 = WMMA4(ACr[j], Bi[j], m3);
        m4 = WMMA4(ACi[j], Br[j], m4);
      }
      Or = m1 - m2; Oi = m3 + m4;
    }

    if (l < 3) {                                 // stage for next layer
      #pragma unroll
      for (int v = 0; v < 8; ++v) {
        L[wb + v]       = Or[v];
        L[256 + wb + v] = Oi[v];
      }
    }
  }

  // ---- readout: Z expectations straight from registers -------------------
  // Registers hold O^T: row c''=v+8s, col r=n; true col c = c'' ^ (15 if r odd)
  float pv[8], P = 0.f;
  #pragma unroll
  for (int v = 0; v < 8; ++v) {
    const float p = Or[v] * Or[v] + Oi[v] * Oi[v];
    pv[v] = p; P += p;
  }
  float e[8];
  #pragma unroll
  for (int q = 0; q < 4; ++q) e[q] = ((n >> (3 - q)) & 1) ? -P : P;
  e[4] = e[5] = e[6] = e[7] = 0.f;
  const int flip = (n & 1) ? 15 : 0;
  #pragma unroll
  for (int v = 0; v < 8; ++v) {
    const int c = (v + 8 * s) ^ flip;
    e[4] += ((c >> 3) & 1) ? -pv[v] : pv[v];
    e[5] += ((c >> 2) & 1) ? -pv[v] : pv[v];
    e[6] += ((c >> 1) & 1) ? -pv[v] : pv[v];
    e[7] += ((c     ) & 1) ? -pv[v] : pv[v];
  }
  #pragma unroll
  for (int k = 0; k < 8; ++k) {
    float v = e[k];
    v += __shfl_xor(v, 1, 32);
    v += __shfl_xor(v, 2, 32);
    v += __shfl_xor(v, 4, 32);
    v += __shfl_xor(v, 8, 32);
    v += __shfl_xor(v, 16, 32);
    e[k] = v;
  }
  if (lane == 0) {
    float4* o4 = (float4*)(out + (size_t)b * 8);
    o4[0] = make_float4(e[0], e[1], e[2], e[3]);
    o4[1] = make_float4(e[4], e[5], e[6], e[7]);
  }
}

extern "C" void kernel_launch(void* const* d_in, const int* in_sizes, int n_in,
                              void* d_out, int out_size, void* d_ws, size_t ws_size,
                              hipStream_t stream) {
  const float* x = (const float*)d_in[0];   // (B, 8) f32
  const float* w = (const float*)d_in[1];   // (4, 8, 2) f32
  float* out = (float*)d_out;               // (B, 8) f32
  float* gws = (float*)d_ws;                // 16 KB gate constants
  const int B = in_sizes[0] / 8;

  qprep_gates_81458349<<<1, 256, 0, stream>>>(w, gws);
  const int blocks = (B + 7) / 8;           // 8 waves (batch elems) per block
  qsim_wmma_81458349<<<blocks, 256, 0, stream>>>(x, gws, out, B);
}